// SelfOrganizingMap_3813930958891
// MI455X (gfx1250) — compile-verified
//
#include <hip/hip_runtime.h>
#include <math.h>

typedef float v2f __attribute__((ext_vector_type(2)));
typedef float v8f __attribute__((ext_vector_type(8)));

namespace {
constexpr int kDim = 512;     // DIM
constexpr int kNs  = 4096;    // N_SAMPLES
constexpr int kNw  = 16384;   // M * N_GRID
constexpr int kChunk = 128;   // K-chunk staged in LDS
constexpr int kPitch = 132;   // LDS row pitch for [sample][k] chunk (pad 4)
// ws layout (float slots)
constexpr int XN_OFF   = 0;                 // x_norm[4096]
constexpr int WN_OFF   = 4096;              // w_norm[16384]
constexpr int D2_OFF   = WN_OFF + kNw;      // dists^2 accum [16384]
constexpr int XL_OFF   = D2_OFF + kNw;      // x[:, -1] compacted [512]
constexpr int META_OFF = XL_OFF + kDim;     // int meta: bmu, update, new_output
constexpr float kBeta = 0.15f;
constexpr float kSigmaOp2 = (64.0f * 0.15f) * (64.0f * 0.15f); // (SIGMA*BETA)^2
}

#if defined(__gfx1250__) && __has_builtin(__builtin_amdgcn_global_load_async_to_lds_b32)
#define SOM_HAVE_ASYNC_LDS 1
#else
#define SOM_HAVE_ASYNC_LDS 0
#endif

// Per-lane global->LDS 4-byte copy (scatter-capable on both sides).
__device__ __forceinline__ void som_copy_g2l_f32(const float* g, float* l) {
#if SOM_HAVE_ASYNC_LDS
  __attribute__((address_space(1))) int* gp =
      (__attribute__((address_space(1))) int*)(void*)const_cast<float*>(g);
  __attribute__((address_space(3))) int* lp =
      (__attribute__((address_space(3))) int*)(void*)l;
  __builtin_amdgcn_global_load_async_to_lds_b32(gp, lp, /*offset=*/0, /*cpol=*/0);
#else
  *l = *g;
#endif
}

__device__ __forceinline__ void som_wait_async() {
#if SOM_HAVE_ASYNC_LDS
#if __has_builtin(__builtin_amdgcn_s_wait_asynccnt)
  __builtin_amdgcn_s_wait_asynccnt(0);
#else
  asm volatile("s_wait_asynccnt 0x0" ::: "memory");
#endif
#endif
}

// ---- per-sample squared norm of x columns; also compact x[:, NS-1] ----
__global__ __launch_bounds__(256) void som_xnorm_kernel(
    const float* __restrict__ x, float* __restrict__ ws) {
  const int i = blockIdx.x * blockDim.x + threadIdx.x;  // 0..4095
  float s = 0.f;
  for (int d = 0; d < kDim; ++d) {
    const float v = x[(size_t)d * kNs + i];
    s += v * v;
  }
  ws[XN_OFF + i] = s;
  if (i < kDim) ws[XL_OFF + i] = x[(size_t)i * kNs + (kNs - 1)];
}

// ---- per-row squared norm of weights; zero-init distance accumulator ----
__global__ __launch_bounds__(256) void som_wnorm_kernel(
    const float* __restrict__ w, float* __restrict__ ws) {
  const int j = blockIdx.x * blockDim.x + threadIdx.x;  // 0..16383
  const float4* wr = (const float4*)(w + (size_t)j * kDim);
  float s = 0.f;
  for (int t = 0; t < kDim / 4; ++t) {
    const float4 v = wr[t];
    s += v.x * v.x + v.y * v.y + v.z * v.z + v.w * v.w;
  }
  ws[WN_OFF + j] = s;
  ws[D2_OFF + j] = 0.f;
}

// ---- main WMMA kernel: sum_i clip(||x_i||^2 + ||w_j||^2 - 2 x_i.w_j) ----
// grid(128, 8) x 256 threads. WG covers 128 weight columns (8 waves x 16).
// blockIdx.y selects 8 of the 64 sample groups (64 samples each).
// Each wave keeps 4 accumulator tiles (4 x 16 samples) so every B fragment
// (weights) load feeds 4 WMMAs -> 4x less L2 traffic on the dominant stream.
__global__ __launch_bounds__(256) void som_gemm_kernel(
    const float* __restrict__ x, const float* __restrict__ w,
    float* __restrict__ ws) {
  __shared__ float lds_x[64 * kPitch];  // transposed x chunk [sample][k]
  __shared__ float lds_xn[64];

  const int tid  = threadIdx.x;
  const int lane = tid & 31;
  const int wave = tid >> 5;
  const int half = lane >> 4;     // lane half selects K pair
  const int n    = lane & 15;     // N (column) / M (row) index within tile
  const int koff = half * 2;

  const int jcol = blockIdx.x * 128 + wave * 16 + n;
  const float wnj = ws[WN_OFF + jcol];
  const float* wrow = w + (size_t)jcol * kDim + koff;  // B row, K-pair offset

  const int ls  = tid & 63;   // sample index for cooperative staging
  const int lk0 = tid >> 6;   // starting k (4 k-rows per pass, 256 threads)

  float colsum = 0.f;
  const int g0 = blockIdx.y * 8;
  for (int g = g0; g < g0 + 8; ++g) {
    const int ibase = g * 64;
    const v8f vzero = {0.f, 0.f, 0.f, 0.f, 0.f, 0.f, 0.f, 0.f};
    v8f acc[4];
#pragma unroll
    for (int q = 0; q < 4; ++q) acc[q] = vzero;

    for (int kc = 0; kc < kDim; kc += kChunk) {
      __syncthreads();
      // stage xs[ibase..ibase+63][kc..kc+127] (transpose-gather from x[k][i])
      for (int k = lk0; k < kChunk; k += 4)
        som_copy_g2l_f32(&x[(size_t)(kc + k) * kNs + ibase + ls],
                         &lds_x[ls * kPitch + k]);
      if (kc == 0 && tid < 64) lds_xn[tid] = ws[XN_OFF + ibase + tid];
      som_wait_async();
      __syncthreads();

      const float* bp = wrow + kc;
#pragma unroll 4
      for (int k0 = 0; k0 < kChunk; k0 += 4) {
        const v2f b = *(const v2f*)(bp + k0);
#pragma unroll
        for (int q = 0; q < 4; ++q) {
          const v2f a = *(const v2f*)(&lds_x[(q * 16 + n) * kPitch + k0 + koff]);
          acc[q] = __builtin_amdgcn_wmma_f32_16x16x4_f32(
              false, a, false, b, (short)0, acc[q], false, false);
        }
      }
    }
#pragma unroll
    for (int q = 0; q < 4; ++q) {
#pragma unroll
      for (int r = 0; r < 8; ++r) {
        const float xnv = lds_xn[q * 16 + r + 8 * half];  // row M = r + 8*half
        colsum += fmaxf(0.f, xnv + wnj - 2.f * acc[q][r]);
      }
    }
  }
  colsum += __shfl_xor(colsum, 16);  // combine lane halves (rows 0-7 + 8-15)
  if (lane < 16) atomicAdd(&ws[D2_OFF + jcol], colsum);
}

// ---- argmin over dists^2 + scalar decision logic ----
__global__ __launch_bounds__(256) void som_argmin_kernel(
    const int* __restrict__ outl, const int* __restrict__ lstm_p,
    const int* __restrict__ real_p, float* __restrict__ ws,
    float* __restrict__ out) {
  __shared__ float sv[256];
  __shared__ int si[256];
  const int t = threadIdx.x;
  float bv = 3.4e38f;
  int bidx = kNw;
  for (int j = t; j < kNw; j += 256) {
    const float v = ws[D2_OFF + j];
    if (v < bv || (v == bv && j < bidx)) { bv = v; bidx = j; }
  }
  sv[t] = bv; si[t] = bidx;
  __syncthreads();
  for (int s = 128; s > 0; s >>= 1) {
    if (t < s) {
      const float ov = sv[t + s];
      const int oi = si[t + s];
      if (ov < sv[t] || (ov == sv[t] && oi < si[t])) { sv[t] = ov; si[t] = oi; }
    }
    __syncthreads();
  }
  if (t == 0) {
    const int bmu = si[0];
    const int som = outl[bmu];
    const int lstm = *lstm_p;
    const int real = *real_p;
    const int le = abs(real - lstm);
    const int se = abs(real - som);
    const int mx = le > se ? le : se;
    const bool c1 = (mx == 0);
    const bool c2 = (mx > 0) && (le == 0);
    const bool c3 = (mx > 0) && (se == 0);
    const int penalty = c1 ? -1 : (c2 ? 1 : 0);
    const int newo = (c1 || c2) ? lstm : (c3 ? som : real);
    const int upd = (c1 || (c3 && !c2)) ? 1 : 0;
    out[0] = (float)penalty;
    out[1] = (float)newo;
    int* mi = (int*)&ws[META_OFF];
    mi[0] = bmu; mi[1] = upd; mi[2] = newo;
  }
}

// ---- weight + output-layer update: one block per weight row ----
__global__ __launch_bounds__(128) void som_update_kernel(
    const float* __restrict__ w, const int* __restrict__ loc,
    const int* __restrict__ outl, const float* __restrict__ ws,
    float* __restrict__ out) {
  const int j = blockIdx.x;
  const int* mi = (const int*)&ws[META_OFF];
  const int bmu = mi[0], upd = mi[1], newo = mi[2];
  const float bi = (float)loc[2 * bmu], bj = (float)loc[2 * bmu + 1];
  const float li = (float)loc[2 * j],   lj = (float)loc[2 * j + 1];
  const float d2 = (li - bi) * (li - bi) + (lj - bj) * (lj - bj);
  const float lr = kBeta * expf(-d2 / kSigmaOp2);
  const float coef = upd ? lr : 0.f;

  const int t = threadIdx.x;  // 128 threads x 4 floats = 512
  const float4 wv = ((const float4*)(w + (size_t)j * kDim))[t];
  const float4 xl = ((const float4*)(ws + XL_OFF))[t];
  float* o = out + 2 + (size_t)j * kDim + t * 4;  // 8B-aligned -> float2 stores
  float2 r0, r1;
  r0.x = wv.x + coef * (xl.x - wv.x);
  r0.y = wv.y + coef * (xl.y - wv.y);
  r1.x = wv.z + coef * (xl.z - wv.z);
  r1.y = wv.w + coef * (xl.w - wv.w);
  *(float2*)o = r0;
  *(float2*)(o + 2) = r1;
  if (t == 0)
    out[2 + (size_t)kNw * kDim + j] = (j == bmu) ? (float)newo : (float)outl[j];
}

extern "C" void kernel_launch(void* const* d_in, const int* in_sizes, int n_in,
                              void* d_out, int out_size, void* d_ws, size_t ws_size,
                              hipStream_t stream) {
  const float* x    = (const float*)d_in[0];  // (512, 4096)
  const float* w    = (const float*)d_in[1];  // (16384, 512)
  const int*   loc  = (const int*)d_in[2];    // (16384, 2)
  const int*   outl = (const int*)d_in[3];    // (16384,)
  const int*   lstm = (const int*)d_in[4];
  const int*   real = (const int*)d_in[5];
  float* out = (float*)d_out;
  float* ws  = (float*)d_ws;

  som_xnorm_kernel<<<kNs / 256, 256, 0, stream>>>(x, ws);
  som_wnorm_kernel<<<kNw / 256, 256, 0, stream>>>(w, ws);
  som_gemm_kernel<<<dim3(128, 8), 256, 0, stream>>>(x, w, ws);
  som_argmin_kernel<<<1, 256, 0, stream>>>(outl, lstm, real, ws, out);
  som_update_kernel<<<kNw, 128, 0, stream>>>(w, loc, outl, ws, out);
}